// CoverageLoss_32401233281614
// MI455X (gfx1250) — compile-verified
//
#include <hip/hip_runtime.h>

// CDNA5 / gfx1250. wave32. One block = 256 threads = 8 waves = 256 samples of
// one primitive (b,p). Grid = B*P*4 = 16384 blocks.
//
// Memory-bound kernel (~193 MB moved -> ~8.3 us at 23.3 TB/s). Bulk global
// traffic is staged through LDS so loads/stores are fully coalesced. Staging
// uses gfx1250 async global<->LDS instructions (ASYNCcnt) when the toolchain
// exposes them; the 3x3 rotation runs on the matrix pipe with
// V_WMMA_F32_16X16X4_F32 (16 samples per WMMA, 2 WMMAs per wave).

typedef __attribute__((ext_vector_type(2))) float v2f;
typedef __attribute__((ext_vector_type(8))) float v8f;

#define NB   64
#define NP   64
#define NS   1024
#define GRID3 32768   // 32^3

// ---- gfx1250 async global<->LDS staging (guarded; falls back to plain) ----
#if defined(__AMDGCN__) && __has_builtin(__builtin_amdgcn_global_load_async_to_lds_b32) && \
    __has_builtin(__builtin_amdgcn_global_store_async_from_lds_b32)
#define HAVE_ASYNC_LDS 1
typedef __attribute__((address_space(1))) int gas_int;   // global (addrspace 1)
typedef __attribute__((address_space(3))) int las_int;   // LDS    (addrspace 3)
__device__ __forceinline__ void stage_g2l(float* l, const float* g) {
    // each lane copies one dword from global address g to LDS address l ( ASYNCcnt-tracked, no VGPR round trip)
    __builtin_amdgcn_global_load_async_to_lds_b32((gas_int*)g, (las_int*)l, 0, 0);
}
__device__ __forceinline__ void stage_l2g(float* g, const float* l) {
    // copies one dword from LDS address l to global address g
    __builtin_amdgcn_global_store_async_from_lds_b32((gas_int*)g, (las_int*)l, 0, 0);
}
__device__ __forceinline__ void async_wait0() {
#if __has_builtin(__builtin_amdgcn_s_wait_asynccnt)
    __builtin_amdgcn_s_wait_asynccnt(0);
#else
    asm volatile("s_wait_asynccnt 0x0" ::: "memory");
#endif
}
#else
#define HAVE_ASYNC_LDS 0
__device__ __forceinline__ void stage_g2l(float* l, const float* g) { *l = *g; }
__device__ __forceinline__ void stage_l2g(float* g, const float* l) { *g = *l; }
__device__ __forceinline__ void async_wait0() {}
#endif

// ---------------- kernel 1: per-primitive weight scalar ----------------
__global__ __launch_bounds__(64)
void cov_weight_kernel(const float* __restrict__ shape,
                       const int* __restrict__ iou,
                       float* __restrict__ wsc)
{
    __shared__ float red[64];
    __shared__ float sSum;
    const int b  = blockIdx.x;
    const int p  = threadIdx.x;
    const int bp = b * NP + p;
    const float sx = shape[bp * 3 + 0];
    const float sy = shape[bp * 3 + 1];
    const float sz = shape[bp * 3 + 2];
    const float area = 8.0f * (sx * sy + sy * sz + sz * sx);
    red[p] = area;
    __syncthreads();
    if (p == 0) {
        float s = 0.0f;
        for (int i = 0; i < 64; ++i) s += red[i];
        sSum = fmaxf(s, 1e-8f);
    }
    __syncthreads();
    float w = area / sSum;
    w = (iou[bp] == 1) ? (w * (1.0f / (float)NS)) : 0.0f;
    wsc[bp] = w;
}

// ---------------- kernel 2: main sample/rotate/gather kernel ----------------
__global__ __launch_bounds__(256)
void cov_main_kernel(const float* __restrict__ shape,
                     const float* __restrict__ trans,
                     const float* __restrict__ quat,
                     const float* __restrict__ CP,
                     const int*   __restrict__ iou,
                     const float* __restrict__ usamp,
                     const float* __restrict__ wsc,
                     float* __restrict__ out)
{
    __shared__ float sR[9];      // rotation matrix (row major, pts_i = R[i][j]*loc_j)
    __shared__ float sSh[3];     // half-dims
    __shared__ float sTr[3];     // translation
    __shared__ float sWv;        // weight scalar
    __shared__ int   sIo;        // iou flag
    __shared__ float sIn[768];   // staging: unit samples -> (later) CPlist
    __shared__ float sLoc[768];  // local (scaled surface) coords
    __shared__ float sPts[768];  // rotated+translated points (WMMA output)

    const int tid   = threadIdx.x;
    const int blk   = blockIdx.x;
    const int chunk = blk & 3;          // 4 blocks per primitive
    const int bp    = blk >> 2;         // 0..4095
    const int b     = bp >> 6;
    const int sBase = chunk << 8;       // sample base (0,256,512,768)

    // float-flat base of this block's 768-float tile inside [B,P,S,3]
    const long long gBase = ((long long)bp * NS + sBase) * 3;

    // async coalesced tile load of unit samples into LDS
    stage_g2l(&sIn[tid      ], &usamp[gBase + tid      ]);
    stage_g2l(&sIn[tid + 256], &usamp[gBase + tid + 256]);
    stage_g2l(&sIn[tid + 512], &usamp[gBase + tid + 512]);

    if (tid == 0) {
        float qw = quat[bp * 4 + 0], qx = quat[bp * 4 + 1];
        float qy = quat[bp * 4 + 2], qz = quat[bp * 4 + 3];
        const float inv = 1.0f / sqrtf(qw * qw + qx * qx + qy * qy + qz * qz);
        qw *= inv; qx *= inv; qy *= inv; qz *= inv;
        sR[0] = 1.0f - 2.0f * (qy * qy + qz * qz);
        sR[1] = 2.0f * (qx * qy - qw * qz);
        sR[2] = 2.0f * (qx * qz + qw * qy);
        sR[3] = 2.0f * (qx * qy + qw * qz);
        sR[4] = 1.0f - 2.0f * (qx * qx + qz * qz);
        sR[5] = 2.0f * (qy * qz - qw * qx);
        sR[6] = 2.0f * (qx * qz - qw * qy);
        sR[7] = 2.0f * (qy * qz + qw * qx);
        sR[8] = 1.0f - 2.0f * (qx * qx + qy * qy);
        sSh[0] = shape[bp * 3 + 0]; sSh[1] = shape[bp * 3 + 1]; sSh[2] = shape[bp * 3 + 2];
        sTr[0] = trans[bp * 3 + 0]; sTr[1] = trans[bp * 3 + 1]; sTr[2] = trans[bp * 3 + 2];
        sWv = wsc[bp];
        sIo = iou[bp];
    }

    async_wait0();     // this wave's async LDS writes are committed
    __syncthreads();   // sIn ready block-wide; sR/sSh/sTr ready

    // project onto unit-cube surface, scale by half-dims
    {
        const float ux = sIn[tid * 3 + 0] - 0.5f;
        const float uy = sIn[tid * 3 + 1] - 0.5f;
        const float uz = sIn[tid * 3 + 2] - 0.5f;
        const float m  = fmaxf(fabsf(ux), fmaxf(fabsf(uy), fabsf(uz)));
        const float iv = 0.5f / fmaxf(m, 1e-6f);
        sLoc[tid * 3 + 0] = ux * iv * sSh[0];
        sLoc[tid * 3 + 1] = uy * iv * sSh[1];
        sLoc[tid * 3 + 2] = uz * iv * sSh[2];
    }
    __syncthreads();   // sLoc ready for cross-lane A-matrix build

    // ---- rotation via V_WMMA_F32_16X16X4_F32: D[s,i] = sum_k loc[s,k]*R[i][k] + t[i]
    const int  lane     = tid & 31;
    const int  waveBase = tid & ~31;
    const int  n        = lane & 15;     // column index (output coord for n<3)
    const bool hiHalf   = lane >= 16;

    // B (4x16): B[K][n] = R[n][K]; lanes 0-15 hold rows K=0,1; lanes 16-31 K=2,3
    float b0 = 0.0f, b1 = 0.0f;
    if (n < 3) {
        b0 = hiHalf ? sR[n * 3 + 2] : sR[n * 3 + 0];
        b1 = hiHalf ? 0.0f          : sR[n * 3 + 1];
    }
    const v2f Bm = {b0, b1};

    // C (16x16): every row = translation broadcast over valid columns
    const float cval = (n < 3) ? sTr[n] : 0.0f;
    const v8f  Cm = {cval, cval, cval, cval, cval, cval, cval, cval};

    for (int t = 0; t < 2; ++t) {        // 2 tiles of 16 samples per wave
        // A (16x4): lane holds row M=lane&15; lanes 0-15 carry K=0,1; 16-31 carry K=2,3
        const int   sIdx = waveBase + t * 16 + n;
        const float a0 = hiHalf ? sLoc[sIdx * 3 + 2] : sLoc[sIdx * 3 + 0];
        const float a1 = hiHalf ? 0.0f               : sLoc[sIdx * 3 + 1];
        const v2f Am = {a0, a1};

        v8f D = __builtin_amdgcn_wmma_f32_16x16x4_f32(
            false, Am, false, Bm, (short)0, Cm, false, false);

        // D element j in lane = D[M = j + (hi?8:0), N = n]; keep columns n<3
        if (n < 3) {
            const int mbase = waveBase + t * 16 + (hiHalf ? 8 : 0);
            #pragma unroll
            for (int j = 0; j < 8; ++j)
                sPts[(mbase + j) * 3 + n] = D[j];
        }
    }
    __syncthreads();   // sPts ready

    // voxel gather (CP is 25 MB total -> L2 resident) + masked select
    const float px = sPts[tid * 3 + 0];
    const float py = sPts[tid * 3 + 1];
    const float pz = sPts[tid * 3 + 2];
    int ix = (int)((px + 0.5f) * 32.0f); ix = ix < 0 ? 0 : (ix > 31 ? 31 : ix);
    int iy = (int)((py + 0.5f) * 32.0f); iy = iy < 0 ? 0 : (iy > 31 ? 31 : iy);
    int iz = (int)((pz + 0.5f) * 32.0f); iz = iz < 0 ? 0 : (iz > 31 ? 31 : iz);
    const int lin = (ix * 32 + iy) * 32 + iz;
    const float* cp3 = CP + ((long long)b * GRID3 + lin) * 3;
    const float gx = cp3[0], gy = cp3[1], gz = cp3[2];
    const bool msk = (sIo == 1);
    sIn[tid * 3 + 0] = msk ? gx : px;    // reuse sIn as CPlist staging
    sIn[tid * 3 + 1] = msk ? gy : py;
    sIn[tid * 3 + 2] = msk ? gz : pz;
    __syncthreads();   // staged CPlist + sPts stable block-wide

    // coalesced stores: pts & CPlist async straight out of LDS; weight direct
    const long long OFF_W = (long long)NB * NP * NS * 3;          // 12582912
    const long long OFF_C = OFF_W + (long long)NB * NP * NS;       // 16777216
    stage_l2g(&out[gBase + tid      ], &sPts[tid      ]);
    stage_l2g(&out[gBase + tid + 256], &sPts[tid + 256]);
    stage_l2g(&out[gBase + tid + 512], &sPts[tid + 512]);
    out[OFF_W + (long long)bp * NS + sBase + tid] = sWv;
    stage_l2g(&out[OFF_C + gBase + tid      ], &sIn[tid      ]);
    stage_l2g(&out[OFF_C + gBase + tid + 256], &sIn[tid + 256]);
    stage_l2g(&out[OFF_C + gBase + tid + 512], &sIn[tid + 512]);
    async_wait0();     // drain async stores (S_ENDPGM also implies wait-idle)
}

// ---------------- launcher ----------------
extern "C" void kernel_launch(void* const* d_in, const int* in_sizes, int n_in,
                              void* d_out, int out_size, void* d_ws, size_t ws_size,
                              hipStream_t stream)
{
    const float* shape = (const float*)d_in[0];   // [B,P,3]
    const float* trans = (const float*)d_in[1];   // [B,P,3]
    const float* quat  = (const float*)d_in[2];   // [B,P,4]
    const float* CP    = (const float*)d_in[3];   // [B,32,32,32,3]
    const int*   iouP  = (const int*)  d_in[4];   // [B,P]
    const float* us    = (const float*)d_in[5];   // [B,P,S,3]
    float* out = (float*)d_out;
    float* wsc = (float*)d_ws;                    // 4096 floats of scratch

    cov_weight_kernel<<<NB, NP, 0, stream>>>(shape, iouP, wsc);
    cov_main_kernel<<<NB * NP * 4, 256, 0, stream>>>(shape, trans, quat, CP,
                                                     iouP, us, wsc, out);
}